// S4DModel_18305150616245
// MI455X (gfx1250) — compile-verified
//
#include <hip/hip_runtime.h>
#include <hip/hip_bf16.h>

// ---------------------------------------------------------------------------
// S4D forward for MI455X (gfx1250, wave32).
// FFN GEMMs use V_WMMA_F32_16X16X4_F32 (full fp32 precision, matches ref)
// with 32x64 register blocking per wave: 8 WMMAs per 6 b64 loads.
// S4D block = exact linear recurrence (equivalent to the reference's
// zero-padded rfft linear convolution), one wave32 per (b,h) channel.
// ---------------------------------------------------------------------------

#define NLAYER 4
#define Hdim   256
#define Nstate 64
#define CIN    64
#define COUT   8
#define Bdim   8
#define Lseq   1024
#define LN_EPS 1e-5f

typedef float v2f __attribute__((ext_vector_type(2)));
typedef float v8f __attribute__((ext_vector_type(8)));

// ---------------------------------------------------------------------------
// x[b,l,h] = stim[b,:,l] @ Win[h,:] + b_in[h] + istate[b,:] @ Wst[h,:] + b_st[h]
// ---------------------------------------------------------------------------
__global__ void input_proj_kernel(const float* __restrict__ stim,
                                  const float* __restrict__ istate,
                                  const float* __restrict__ Win,
                                  const float* __restrict__ b_in,
                                  const float* __restrict__ Wst,
                                  const float* __restrict__ b_st,
                                  float* __restrict__ x) {
    int idx = blockIdx.x * blockDim.x + threadIdx.x;          // over B*L*H
    int h  = idx & (Hdim - 1);
    int bl = idx >> 8;                                        // Hdim==256
    int l  = bl & (Lseq - 1);
    int b  = bl >> 10;
    float acc = b_in[h] + b_st[h];
    const float* sp = stim + ((size_t)b * CIN) * Lseq + l;    // stride Lseq over c
    const float* wp = Win + (size_t)h * CIN;
#pragma unroll 8
    for (int c = 0; c < CIN; ++c) acc += sp[(size_t)c * Lseq] * wp[c];
    const float* ip = istate + (size_t)b * COUT;
    const float* vp = Wst + (size_t)h * COUT;
#pragma unroll
    for (int k = 0; k < COUT; ++k) acc += ip[k] * vp[k];
    x[idx] = acc;
}

// ---------------------------------------------------------------------------
// LayerNorm over H=256; one 256-thread block per (b,l) row.
// transposed==1 : write u[b,h,l]  (layout for the s4d scan)
// transposed==0 : write out[b,l,h]
// ---------------------------------------------------------------------------
__global__ void layernorm_kernel(const float* __restrict__ x,
                                 const float* __restrict__ w,
                                 const float* __restrict__ bcoef,
                                 float* __restrict__ out,
                                 int transposed) {
    __shared__ float red[256];
    int row = blockIdx.x;                  // b*L + l
    int tid = threadIdx.x;                 // h
    float v = x[(size_t)row * Hdim + tid];

    red[tid] = v;
    __syncthreads();
    for (int s = 128; s > 0; s >>= 1) {
        if (tid < s) red[tid] += red[tid + s];
        __syncthreads();
    }
    float mean = red[0] * (1.0f / Hdim);
    __syncthreads();
    float d = v - mean;
    red[tid] = d * d;
    __syncthreads();
    for (int s = 128; s > 0; s >>= 1) {
        if (tid < s) red[tid] += red[tid + s];
        __syncthreads();
    }
    float var = red[0] * (1.0f / Hdim);
    float xn = d * rsqrtf(var + LN_EPS) * w[tid] + bcoef[tid];

    if (transposed) {
        int l = row & (Lseq - 1);
        int b = row >> 10;
        out[((size_t)(b * Hdim + tid)) * Lseq + l] = xn;
    } else {
        out[(size_t)row * Hdim + tid] = xn;
    }
}

// ---------------------------------------------------------------------------
// S4D scan: one wave32 per (b,h). Lane owns states n=lane and n=lane+32.
//   s_n[l] = e_n * s_n[l-1] + u[l],   y[l] = Re( sum_n Cbar_n * s_n[l] ) + D*u[l]
// x[b,l,h] += y[l]
// ---------------------------------------------------------------------------
__global__ void __launch_bounds__(32)
s4d_scan_kernel(const float* __restrict__ u,
                const float* __restrict__ A_re, const float* __restrict__ A_im,
                const float* __restrict__ C_re, const float* __restrict__ C_im,
                const float* __restrict__ D,    const float* __restrict__ log_step,
                float* __restrict__ x) {
    int bh = blockIdx.x;                 // 0..B*H-1
    int b = bh >> 8;
    int h = bh & (Hdim - 1);
    int lane = threadIdx.x;              // 0..31

    float step = __expf(log_step[h]);
    float Dh = D[h];

    float e_r[2], e_i[2], cb_r[2], cb_i[2];
#pragma unroll
    for (int j = 0; j < 2; ++j) {
        int n = lane + 32 * j;
        float ar = A_re[n], ai = A_im[n];
        float dr = step * ar, di = step * ai;
        float em = __expf(dr);
        float er = em * __cosf(di);
        float ei = em * __sinf(di);
        e_r[j] = er; e_i[j] = ei;
        // (e - 1) / A
        float inv = 1.0f / (ar * ar + ai * ai);
        float gr = ((er - 1.0f) * ar + ei * ai) * inv;
        float gi = (ei * ar - (er - 1.0f) * ai) * inv;
        float cr = C_re[(size_t)h * Nstate + n];
        float ci = C_im[(size_t)h * Nstate + n];
        cb_r[j] = cr * gr - ci * gi;
        cb_i[j] = cr * gi + ci * gr;
    }

    float s0r = 0.f, s0i = 0.f, s1r = 0.f, s1i = 0.f;
    const float* up = u + (size_t)bh * Lseq;
    float* xp = x + ((size_t)b * Lseq) * Hdim + h;

    for (int l = 0; l < Lseq; ++l) {
        float uv = up[l];
        float t0r = e_r[0] * s0r - e_i[0] * s0i + uv;
        float t0i = e_r[0] * s0i + e_i[0] * s0r;
        float t1r = e_r[1] * s1r - e_i[1] * s1i + uv;
        float t1i = e_r[1] * s1i + e_i[1] * s1r;
        s0r = t0r; s0i = t0i; s1r = t1r; s1i = t1i;
        float p = cb_r[0] * s0r - cb_i[0] * s0i + cb_r[1] * s1r - cb_i[1] * s1i;
#pragma unroll
        for (int off = 16; off > 0; off >>= 1) p += __shfl_xor(p, off, 32);
        if (lane == 0) xp[(size_t)l * Hdim] += p + Dh * uv;
    }
}

// ---------------------------------------------------------------------------
// WMMA fp32 GEMM:  out[M,Nout] = act( A[M,K] @ W[Nout,K]^T + bias ) (+ residual)
// Register-blocked: one wave computes a 32x64 output tile (2 M-subtiles x
// 4 N-subtiles of 16x16), so each K-step of 4 issues 8 WMMAs on 6 b64 loads.
// A-frag (16x4 f32): lane L -> A[m0+(L&15), k + 2*(L>>4) + {0,1}]
// B-frag (4x16 f32): lane L -> W[n0+(L&15), k + 2*(L>>4) + {0,1}]  (B = W^T)
// C/D  (16x16 f32): vgpr r, lane L -> row r + 8*(L>>4), col (L&15)
// ---------------------------------------------------------------------------
#define MT 2   // 16-row subtiles per wave
#define NT 4   // 16-col subtiles per wave

__global__ void gemm_wmma_f32(const float* __restrict__ A,
                              const float* __restrict__ W,
                              const float* __restrict__ bias,
                              const float* __restrict__ residual,
                              float* __restrict__ out,
                              int M, int K, int Nout, int fuse_gelu) {
    int wave = (blockIdx.x * blockDim.x + threadIdx.x) >> 5;
    int lane = threadIdx.x & 31;
    int ntiles = Nout >> 6;               // 64-wide macro tiles
    int mt = wave / ntiles;
    int nt = wave - mt * ntiles;
    int m0 = mt * (16 * MT), n0 = nt * (16 * NT);

    int half  = lane >> 4;      // 0: K={0,1}, 1: K={2,3} of each 4-step
    int idx16 = lane & 15;

    const float* ap[MT];
    const float* bp[NT];
#pragma unroll
    for (int mi = 0; mi < MT; ++mi)
        ap[mi] = A + (size_t)(m0 + 16 * mi + idx16) * K + 2 * half;
#pragma unroll
    for (int ni = 0; ni < NT; ++ni)
        bp[ni] = W + (size_t)(n0 + 16 * ni + idx16) * K + 2 * half;

    v8f c[MT][NT];
#pragma unroll
    for (int mi = 0; mi < MT; ++mi)
#pragma unroll
        for (int ni = 0; ni < NT; ++ni)
            c[mi][ni] = (v8f){0.f, 0.f, 0.f, 0.f, 0.f, 0.f, 0.f, 0.f};

#pragma unroll 2
    for (int k = 0; k < K; k += 4) {
        v2f a[MT], b[NT];
#pragma unroll
        for (int mi = 0; mi < MT; ++mi) a[mi] = *(const v2f*)(ap[mi] + k);
#pragma unroll
        for (int ni = 0; ni < NT; ++ni) b[ni] = *(const v2f*)(bp[ni] + k);
#pragma unroll
        for (int mi = 0; mi < MT; ++mi)
#pragma unroll
            for (int ni = 0; ni < NT; ++ni)
                c[mi][ni] = __builtin_amdgcn_wmma_f32_16x16x4_f32(
                    false, a[mi], false, b[ni], (short)0, c[mi][ni],
                    false, false);
    }

#pragma unroll
    for (int mi = 0; mi < MT; ++mi) {
#pragma unroll
        for (int ni = 0; ni < NT; ++ni) {
            int col = n0 + 16 * ni + idx16;
            float bv = bias[col];
#pragma unroll
            for (int r = 0; r < 8; ++r) {
                int row = m0 + 16 * mi + r + 8 * half;
                float v = c[mi][ni][r] + bv;
                if (fuse_gelu)
                    v = 0.5f * v * (1.0f + erff(v * 0.70710678118654752f));
                size_t o = (size_t)row * Nout + col;
                if (residual) v += residual[o];
                out[o] = v;
            }
        }
    }
}

// ---------------------------------------------------------------------------
// out[b,o,l] = x[b,l,:] @ Wout[o,:] + b_out[o]
// ---------------------------------------------------------------------------
__global__ void out_proj_kernel(const float* __restrict__ x,
                                const float* __restrict__ Wout,
                                const float* __restrict__ b_out,
                                float* __restrict__ out) {
    int idx = blockIdx.x * blockDim.x + threadIdx.x;   // over B*COUT*L, l fastest
    int l = idx & (Lseq - 1);
    int t = idx >> 10;
    int o = t & (COUT - 1);
    int b = t >> 3;
    float acc = b_out[o];
    const float* xr = x + ((size_t)(b * Lseq + l)) * Hdim;
    const float* wr = Wout + (size_t)o * Hdim;
#pragma unroll 8
    for (int h = 0; h < Hdim; ++h) acc += xr[h] * wr[h];
    out[idx] = acc;
}

// ---------------------------------------------------------------------------
extern "C" void kernel_launch(void* const* d_in, const int* in_sizes, int n_in,
                              void* d_out, int out_size, void* d_ws, size_t ws_size,
                              hipStream_t stream) {
    const float* stim     = (const float*)d_in[0];
    const float* istate   = (const float*)d_in[1];
    const float* A_re     = (const float*)d_in[2];
    const float* A_im     = (const float*)d_in[3];
    const float* C_re     = (const float*)d_in[4];
    const float* C_im     = (const float*)d_in[5];
    const float* Dp       = (const float*)d_in[6];
    const float* log_step = (const float*)d_in[7];
    const float* ln1_w    = (const float*)d_in[8];
    const float* ln1_b    = (const float*)d_in[9];
    const float* ln2_w    = (const float*)d_in[10];
    const float* ln2_b    = (const float*)d_in[11];
    const float* W1       = (const float*)d_in[12];
    const float* b1       = (const float*)d_in[13];
    const float* W2       = (const float*)d_in[14];
    const float* b2       = (const float*)d_in[15];
    const float* Win      = (const float*)d_in[16];
    const float* b_in     = (const float*)d_in[17];
    const float* Wst      = (const float*)d_in[18];
    const float* b_st     = (const float*)d_in[19];
    const float* Wout     = (const float*)d_in[20];
    const float* b_out    = (const float*)d_in[21];
    float* out = (float*)d_out;

    const size_t ROWS = (size_t)Bdim * Lseq;            // 8192
    float* x    = (float*)d_ws;                         // [ROWS, H]    8 MB
    float* u    = x    + ROWS * Hdim;                   // [B*H, L]     8 MB
    float* xn2  = u    + ROWS * Hdim;                   // [ROWS, H]    8 MB
    float* hbuf = xn2  + ROWS * Hdim;                   // [ROWS, 2H]  16 MB

    // 1) input projection + state fusion
    input_proj_kernel<<<(Bdim * Lseq * Hdim) / 256, 256, 0, stream>>>(
        stim, istate, Win, b_in, Wst, b_st, x);

    // 2) layers
    for (int i = 0; i < NLAYER; ++i) {
        // LN1 -> transposed u[b,h,l]
        layernorm_kernel<<<ROWS, 256, 0, stream>>>(
            x, ln1_w + i * Hdim, ln1_b + i * Hdim, u, 1);

        // S4D scan, x += y
        s4d_scan_kernel<<<Bdim * Hdim, 32, 0, stream>>>(
            u, A_re + i * Nstate, A_im + i * Nstate,
            C_re + (size_t)i * Hdim * Nstate, C_im + (size_t)i * Hdim * Nstate,
            Dp + i * Hdim, log_step + i * Hdim, x);

        // LN2 -> xn2[b,l,h]
        layernorm_kernel<<<ROWS, 256, 0, stream>>>(
            x, ln2_w + i * Hdim, ln2_b + i * Hdim, xn2, 0);

        // FFN up: hbuf = gelu(xn2 @ W1^T + b1)   [8192,256]x[256,512]
        {
            int M = (int)ROWS, K = Hdim, Nout = 2 * Hdim;
            int waves = (M / (16 * MT)) * (Nout / (16 * NT));   // 256*8 = 2048
            gemm_wmma_f32<<<waves * 32 / 256, 256, 0, stream>>>(
                xn2, W1 + (size_t)i * 2 * Hdim * Hdim, b1 + i * 2 * Hdim,
                nullptr, hbuf, M, K, Nout, 1);
        }
        // FFN down + residual: x += hbuf @ W2^T + b2   [8192,512]x[512,256]
        {
            int M = (int)ROWS, K = 2 * Hdim, Nout = Hdim;
            int waves = (M / (16 * MT)) * (Nout / (16 * NT));   // 256*4 = 1024
            gemm_wmma_f32<<<waves * 32 / 256, 256, 0, stream>>>(
                hbuf, W2 + (size_t)i * Hdim * 2 * Hdim, b2 + i * Hdim,
                x, x, M, K, Nout, 0);
        }
    }

    // 3) output projection -> [B, COUT, L]
    out_proj_kernel<<<(Bdim * COUT * Lseq) / 256, 256, 0, stream>>>(
        x, Wout, b_out, out);
}